// NemotronHDecoderLayer_12360915878724
// MI455X (gfx1250) — compile-verified
//
#include <hip/hip_runtime.h>
#include <math.h>

#define SDIM 256
#define HDIM 4096
#define NHEAD 128
#define PDIM 64
#define GDIM 8
#define NDIM 128
#define KCONV 4
#define DINNER 8192
#define DSTATE 1024
#define CONVDIM 10240
#define INOUT 18560
#define EPSV 1e-6f

typedef __attribute__((ext_vector_type(2))) float v2f;
typedef __attribute__((ext_vector_type(8))) float v8f;

__device__ __forceinline__ float sigmoidf_(float x) { return 1.0f / (1.0f + expf(-x)); }

__device__ __forceinline__ float block_reduce_sum(float v) {
    __shared__ float sw[8];
    const int lane = threadIdx.x & 31;
    const int wv = threadIdx.x >> 5;
    #pragma unroll
    for (int o = 16; o > 0; o >>= 1) v += __shfl_down(v, o, 32);
    if (lane == 0) sw[wv] = v;
    __syncthreads();
    if (wv == 0) {
        v = (lane < (int)(blockDim.x >> 5)) ? sw[lane] : 0.0f;
        #pragma unroll
        for (int o = 4; o > 0; o >>= 1) v += __shfl_down(v, o, 32);
        if (lane == 0) sw[0] = v;
    }
    __syncthreads();
    v = sw[0];
    __syncthreads();
    return v;
}

// -------- 1) RMSNorm of hidden_states --------
__global__ void __launch_bounds__(256) rmsnorm_kernel(const float* __restrict__ x,
                                                      const float* __restrict__ w,
                                                      float* __restrict__ out) {
    const int s = blockIdx.x;
    const float* row = x + (size_t)s * HDIM;
    float ss = 0.0f;
    for (int i = threadIdx.x; i < HDIM; i += 256) { float v = row[i]; ss += v * v; }
    ss = block_reduce_sum(ss);
    const float inv = rsqrtf(ss / (float)HDIM + EPSV);
    float* orow = out + (size_t)s * HDIM;
    for (int i = threadIdx.x; i < HDIM; i += 256) orow[i] = row[i] * inv * w[i];
}

// -------- 2) in-proj GEMM: proj[S][INOUT] = h[S][HDIM] * w_in[INOUT][HDIM]^T --------
// grid.x = INOUT/16 = 1160 ; 128 threads = 4 waves ; wave owns 64 M-rows (4 tiles)
__global__ void __launch_bounds__(128) inproj_kernel(const float* __restrict__ A,
                                                     const float* __restrict__ W,
                                                     float* __restrict__ P) {
    const int n0 = blockIdx.x * 16;
    const int wave = threadIdx.x >> 5;
    const int lane = threadIdx.x & 31;
    const int half = lane >> 4;
    const int r = lane & 15;
    const int mb = wave * 64;

    const float* wrow = W + (size_t)(n0 + r) * HDIM + 2 * half;
    const float* a0 = A + (size_t)(mb + 0  + r) * HDIM + 2 * half;
    const float* a1 = A + (size_t)(mb + 16 + r) * HDIM + 2 * half;
    const float* a2 = A + (size_t)(mb + 32 + r) * HDIM + 2 * half;
    const float* a3 = A + (size_t)(mb + 48 + r) * HDIM + 2 * half;

    v8f c0 = {}, c1 = {}, c2 = {}, c3 = {};
    for (int kb = 0; kb < HDIM; kb += 32) {
        __builtin_prefetch(wrow + kb + 512, 0, 1);   // stream weights: global_prefetch_b8
        #pragma unroll
        for (int kk = 0; kk < 32; kk += 4) {
            const int k = kb + kk;
            const v2f b  = *(const v2f*)(wrow + k);
            const v2f x0 = *(const v2f*)(a0 + k);
            const v2f x1 = *(const v2f*)(a1 + k);
            const v2f x2 = *(const v2f*)(a2 + k);
            const v2f x3 = *(const v2f*)(a3 + k);
            c0 = __builtin_amdgcn_wmma_f32_16x16x4_f32(false, x0, false, b, (short)0, c0, false, false);
            c1 = __builtin_amdgcn_wmma_f32_16x16x4_f32(false, x1, false, b, (short)0, c1, false, false);
            c2 = __builtin_amdgcn_wmma_f32_16x16x4_f32(false, x2, false, b, (short)0, c2, false, false);
            c3 = __builtin_amdgcn_wmma_f32_16x16x4_f32(false, x3, false, b, (short)0, c3, false, false);
        }
    }
    const int col = n0 + r;
    #pragma unroll
    for (int v = 0; v < 8; ++v) {
        const int ro = v + 8 * half;
        P[(size_t)(mb + 0  + ro) * INOUT + col] = c0[v];
        P[(size_t)(mb + 16 + ro) * INOUT + col] = c1[v];
        P[(size_t)(mb + 32 + ro) * INOUT + col] = c2[v];
        P[(size_t)(mb + 48 + ro) * INOUT + col] = c3[v];
    }
}

// -------- 3) causal conv1d(K=4) + SiLU ; also emit conv_state_out --------
__global__ void __launch_bounds__(256) conv_silu_kernel(const float* __restrict__ proj,
                                                        const float* __restrict__ cs,
                                                        const float* __restrict__ cw,
                                                        const float* __restrict__ cb,
                                                        float* __restrict__ y,
                                                        float* __restrict__ cs_out) {
    const int idx = blockIdx.x * 256 + threadIdx.x;
    const int c = idx % CONVDIM;
    const int s = idx / CONVDIM;
    if (s >= SDIM) return;
    float acc = cb[c];
    #pragma unroll
    for (int j = 0; j < KCONV; ++j) {
        const int t = s + j;   // padded coordinate
        const float xv = (t < KCONV - 1)
            ? cs[c * (KCONV - 1) + t]
            : proj[(size_t)(t - (KCONV - 1)) * INOUT + DINNER + c];
        acc += cw[c * KCONV + j] * xv;
    }
    acc = acc * sigmoidf_(acc);          // SiLU
    y[(size_t)s * CONVDIM + c] = acc;
    if (s < KCONV - 1)
        cs_out[c * (KCONV - 1) + s] =
            proj[(size_t)(SDIM - (KCONV - 1) + s) * INOUT + DINNER + c];
}

// -------- 4) sequential SSM scan: one workgroup per head, state in registers --------
__global__ void __launch_bounds__(256) ssm_scan_kernel(const float* __restrict__ y,     // [S][CONVDIM]
                                                       const float* __restrict__ proj,  // dt slice
                                                       const float* __restrict__ st_in, // [NH][P][N]
                                                       const float* __restrict__ A_log,
                                                       const float* __restrict__ D_param,
                                                       const float* __restrict__ dt_bias,
                                                       float* __restrict__ ssm_out,     // [S][DINNER]
                                                       float* __restrict__ st_out) {    // [NH][P][N]
    const int h = blockIdx.x;
    const int tid = threadIdx.x;
    const int p = tid >> 2;            // 0..63
    const int nq = (tid & 3) * 32;     // 32 n-values per thread
    const int g = h >> 4;              // NH/G = 16 heads per group

    float st[32];
    const size_t sbase = (size_t)h * PDIM * NDIM + (size_t)p * NDIM + nq;
    #pragma unroll
    for (int i = 0; i < 32; ++i) st[i] = st_in[sbase + i];

    const float Av  = -expf(A_log[h]);
    const float Dv  = D_param[h];
    const float dtb = dt_bias[h];

    __shared__ float sB[NDIM];
    __shared__ float sC[NDIM];
    __shared__ float sX[PDIM];

    for (int s = 0; s < SDIM; ++s) {
        const size_t yb = (size_t)s * CONVDIM;
        if (tid < 128)      sB[tid]       = y[yb + DINNER + g * NDIM + tid];
        else                sC[tid - 128] = y[yb + DINNER + DSTATE + g * NDIM + (tid - 128)];
        if (tid < 64)       sX[tid]       = y[yb + h * PDIM + tid];
        float dtr = proj[(size_t)s * INOUT + DINNER + CONVDIM + h] + dtb;
        const float dt = (dtr > 20.0f) ? dtr : log1pf(expf(dtr));   // softplus
        const float dA = expf(dt * Av);
        __syncthreads();
        const float xv = sX[p];
        const float dtx = dt * xv;
        float ysum = 0.0f;
        #pragma unroll
        for (int i = 0; i < 32; ++i) {
            st[i] = st[i] * dA + dtx * sB[nq + i];
            ysum += st[i] * sC[nq + i];
        }
        ysum += __shfl_xor(ysum, 1, 32);
        ysum += __shfl_xor(ysum, 2, 32);
        if ((tid & 3) == 0)
            ssm_out[(size_t)s * DINNER + h * PDIM + p] = ysum + Dv * xv;
        __syncthreads();
    }
    #pragma unroll
    for (int i = 0; i < 32; ++i) st_out[sbase + i] = st[i];
}

// -------- 5) gate (SiLU) * group RMSNorm (group size 1024) --------
__global__ void __launch_bounds__(256) gatenorm_kernel(const float* __restrict__ ssm_o,
                                                       const float* __restrict__ proj,  // gate slice
                                                       const float* __restrict__ nw,
                                                       float* __restrict__ normed) {
    const int s = blockIdx.x >> 3;
    const int grp = blockIdx.x & 7;
    const int tid = threadIdx.x;
    float vals[4];
    float ss = 0.0f;
    #pragma unroll
    for (int i = 0; i < 4; ++i) {
        const int d = grp * 1024 + i * 256 + tid;
        float gt = proj[(size_t)s * INOUT + d];
        gt = gt * sigmoidf_(gt);
        const float v = ssm_o[(size_t)s * DINNER + d] * gt;
        vals[i] = v;
        ss += v * v;
    }
    ss = block_reduce_sum(ss);
    const float inv = rsqrtf(ss / 1024.0f + EPSV);
    #pragma unroll
    for (int i = 0; i < 4; ++i) {
        const int d = grp * 1024 + i * 256 + tid;
        normed[(size_t)s * DINNER + d] = vals[i] * inv * nw[d];
    }
}

// -------- 6) out-proj GEMM + residual: out[S][H] = resid + normed * w_out^T --------
// grid.x = HDIM/16 = 256 ; same wave structure as inproj, K = DINNER
__global__ void __launch_bounds__(128) outproj_kernel(const float* __restrict__ A,   // normed [S][DINNER]
                                                      const float* __restrict__ W,   // w_out [HDIM][DINNER]
                                                      const float* __restrict__ resid,
                                                      float* __restrict__ out) {
    const int n0 = blockIdx.x * 16;
    const int wave = threadIdx.x >> 5;
    const int lane = threadIdx.x & 31;
    const int half = lane >> 4;
    const int r = lane & 15;
    const int mb = wave * 64;

    const float* wrow = W + (size_t)(n0 + r) * DINNER + 2 * half;
    const float* a0 = A + (size_t)(mb + 0  + r) * DINNER + 2 * half;
    const float* a1 = A + (size_t)(mb + 16 + r) * DINNER + 2 * half;
    const float* a2 = A + (size_t)(mb + 32 + r) * DINNER + 2 * half;
    const float* a3 = A + (size_t)(mb + 48 + r) * DINNER + 2 * half;

    v8f c0 = {}, c1 = {}, c2 = {}, c3 = {};
    for (int kb = 0; kb < DINNER; kb += 32) {
        __builtin_prefetch(wrow + kb + 512, 0, 1);
        #pragma unroll
        for (int kk = 0; kk < 32; kk += 4) {
            const int k = kb + kk;
            const v2f b  = *(const v2f*)(wrow + k);
            const v2f x0 = *(const v2f*)(a0 + k);
            const v2f x1 = *(const v2f*)(a1 + k);
            const v2f x2 = *(const v2f*)(a2 + k);
            const v2f x3 = *(const v2f*)(a3 + k);
            c0 = __builtin_amdgcn_wmma_f32_16x16x4_f32(false, x0, false, b, (short)0, c0, false, false);
            c1 = __builtin_amdgcn_wmma_f32_16x16x4_f32(false, x1, false, b, (short)0, c1, false, false);
            c2 = __builtin_amdgcn_wmma_f32_16x16x4_f32(false, x2, false, b, (short)0, c2, false, false);
            c3 = __builtin_amdgcn_wmma_f32_16x16x4_f32(false, x3, false, b, (short)0, c3, false, false);
        }
    }
    const int col = n0 + r;
    #pragma unroll
    for (int v = 0; v < 8; ++v) {
        const int ro = v + 8 * half;
        const size_t i0 = (size_t)(mb + 0  + ro) * HDIM + col;
        const size_t i1 = (size_t)(mb + 16 + ro) * HDIM + col;
        const size_t i2 = (size_t)(mb + 32 + ro) * HDIM + col;
        const size_t i3 = (size_t)(mb + 48 + ro) * HDIM + col;
        out[i0] = resid[i0] + c0[v];
        out[i1] = resid[i1] + c1[v];
        out[i2] = resid[i2] + c2[v];
        out[i3] = resid[i3] + c3[v];
    }
}

extern "C" void kernel_launch(void* const* d_in, const int* in_sizes, int n_in,
                              void* d_out, int out_size, void* d_ws, size_t ws_size,
                              hipStream_t stream) {
    const float* hidden     = (const float*)d_in[0];
    const float* conv_state = (const float*)d_in[1];
    const float* ssm_state  = (const float*)d_in[2];
    const float* norm_w     = (const float*)d_in[3];
    const float* w_in       = (const float*)d_in[4];
    const float* conv_w     = (const float*)d_in[5];
    const float* conv_b     = (const float*)d_in[6];
    const float* A_log      = (const float*)d_in[7];
    const float* D_param    = (const float*)d_in[8];
    const float* dt_bias    = (const float*)d_in[9];
    const float* ssm_norm_w = (const float*)d_in[10];
    const float* w_out      = (const float*)d_in[11];

    float* out     = (float*)d_out;                       // [S][H]
    float* cs_out  = out + (size_t)SDIM * HDIM;           // [CONVDIM][K-1]
    float* st_out  = cs_out + (size_t)CONVDIM * (KCONV-1);// [NH][P][N]

    float* ws    = (float*)d_ws;
    float* hbuf  = ws;                                    // [S][H]        4 MB
    float* proj  = hbuf + (size_t)SDIM * HDIM;            // [S][INOUT]   19 MB
    float* yconv = proj + (size_t)SDIM * INOUT;           // [S][CONVDIM] 10.5 MB
    float* ssm_o = yconv + (size_t)SDIM * CONVDIM;        // [S][DINNER]   8.4 MB
    float* normed = yconv;                                // reuse (conv output dead after scan)

    rmsnorm_kernel<<<SDIM, 256, 0, stream>>>(hidden, norm_w, hbuf);
    inproj_kernel<<<INOUT / 16, 128, 0, stream>>>(hbuf, w_in, proj);
    conv_silu_kernel<<<(SDIM * CONVDIM) / 256, 256, 0, stream>>>(proj, conv_state, conv_w, conv_b,
                                                                 yconv, cs_out);
    ssm_scan_kernel<<<NHEAD, 256, 0, stream>>>(yconv, proj, ssm_state, A_log, D_param, dt_bias,
                                               ssm_o, st_out);
    gatenorm_kernel<<<SDIM * GDIM, 256, 0, stream>>>(ssm_o, proj, ssm_norm_w, normed);
    outproj_kernel<<<HDIM / 16, 128, 0, stream>>>(normed, w_out, hidden, out);
}